// CardGNN_15556371546400
// MI455X (gfx1250) — compile-verified
//
#include <hip/hip_runtime.h>
#include <math.h>

// ---------------- problem constants (match reference) ----------------
#define NN      50000          // nodes
#define EE      800000         // raw edges
#define ETOT    (EE + NN)      // edges + self loops
#define INDIM   128
#define HIDC    32             // per-head channels
#define NHEADS  4
#define HC      128            // HIDC*NHEADS
#define EMB     128
#define NEG_SLOPE 0.2f

typedef __attribute__((ext_vector_type(2))) float v2f;
typedef __attribute__((ext_vector_type(8))) float v8f;

// ---------------- helpers ----------------
__device__ __forceinline__ float gelu_exact(float x) {
    return 0.5f * x * (1.0f + erff(x * 0.70710678118654752440f));
}

// monotonic float<->uint mapping so atomicMax(uint) == float max
__device__ __forceinline__ unsigned ordf(float f) {
    unsigned u = __float_as_uint(f);
    return (u >> 31) ? ~u : (u | 0x80000000u);
}
__device__ __forceinline__ float deord(unsigned o) {
    unsigned u = (o >> 31) ? (o ^ 0x80000000u) : ~o;
    return __uint_as_float(u);
}

// ---------------- zero fill ----------------
__global__ void fillz_kernel(float* p, int n) {
    int i = blockIdx.x * blockDim.x + threadIdx.x;
    if (i < n) p[i] = 0.0f;
}

// ---------------- GEMM: C[M,Nout] = act(A[M,K] @ W[K,Nout] + b) ----------------
// block = 32 * (Nout/16) threads; grid.x = M/16 (M must be multiple of 16).
// Each wave computes one 16x16 tile with V_WMMA_F32_16X16X4_F32.
// act: 0 = none, 1 = exact GELU
__global__ void gemm_wmma_kernel(const float* __restrict__ A,
                                 const float* __restrict__ W,
                                 const float* __restrict__ bias,
                                 float* __restrict__ C,
                                 int M, int K, int Nout, int act) {
    const int lane = threadIdx.x & 31;
    const int wave = threadIdx.x >> 5;
    const int hi   = lane >> 4;        // 0 or 1
    const int lo   = lane & 15;
    const int m0   = blockIdx.x * 16;
    const int n0   = wave * 16;

    v8f acc = {};
    const float* arow = A + (size_t)(m0 + lo) * K;   // A fragment: row m0+lo
    for (int k0 = 0; k0 < K; k0 += 4) {
        const int ka = k0 + 2 * hi;
        v2f a, b;
        // A 16x4: VGPR0 = K (lanes0-15: k0, lanes16-31: k0+2); VGPR1 = K+1
        a.x = arow[ka];
        a.y = arow[ka + 1];
        // B 4x16: same K striping, N = lane&15
        b.x = W[(size_t)ka       * Nout + n0 + lo];
        b.y = W[(size_t)(ka + 1) * Nout + n0 + lo];
        acc = __builtin_amdgcn_wmma_f32_16x16x4_f32(
            false, a, false, b, (short)0, acc, false, false);
    }

    const int col = n0 + lo;
    const float bv = bias[col];
#pragma unroll
    for (int v = 0; v < 8; ++v) {
        const int r = m0 + v + 8 * hi;     // C/D layout: VGPR v -> rows v / v+8
        float val = acc[v] + bv;
        if (act == 1) val = gelu_exact(val);
        C[(size_t)r * Nout + col] = val;
    }
}

// ---------------- edge logits + segment max (wave per edge) ----------------
__global__ void edge_logit_kernel(const int* __restrict__ src,
                                  const int* __restrict__ dst,
                                  const float* __restrict__ xl,   // [N,HC] target transform
                                  const float* __restrict__ xr,   // [N,HC] source transform
                                  const float* __restrict__ att,  // [NHEADS,HIDC]
                                  float* __restrict__ logit,      // [ETOT,NHEADS]
                                  unsigned* __restrict__ maxb) {  // [N,NHEADS] ordered-uint
    const int e = blockIdx.x * (blockDim.x >> 5) + (threadIdx.x >> 5);
    if (e >= ETOT) return;
    const int lane = threadIdx.x & 31;
    int s, d;
    if (e < EE) { s = src[e]; d = dst[e]; } else { s = d = e - EE; }

    const float* pl = xl + (size_t)d * HC;
    const float* pr = xr + (size_t)s * HC;
    float p[NHEADS];
#pragma unroll
    for (int h = 0; h < NHEADS; ++h) {
        float v = pl[h * HIDC + lane] + pr[h * HIDC + lane];
        v = (v > 0.0f) ? v : NEG_SLOPE * v;              // leaky_relu
        p[h] = v * att[h * HIDC + lane];
    }
#pragma unroll
    for (int off = 16; off > 0; off >>= 1) {
#pragma unroll
        for (int h = 0; h < NHEADS; ++h)
            p[h] += __shfl_xor(p[h], off, 32);
    }
    if (lane == 0) {
#pragma unroll
        for (int h = 0; h < NHEADS; ++h) {
            logit[(size_t)e * NHEADS + h] = p[h];
            atomicMax(&maxb[(size_t)d * NHEADS + h], ordf(p[h]));
        }
    }
}

// ---------------- exp(logit - max) + segment sum (thread per edge*head) ----------------
__global__ void edge_exp_kernel(const int* __restrict__ dst,
                                const unsigned* __restrict__ maxb,
                                float* __restrict__ ebuf,      // in: logit, out: ex
                                float* __restrict__ den) {     // [N,NHEADS]
    const int idx = blockIdx.x * blockDim.x + threadIdx.x;
    if (idx >= ETOT * NHEADS) return;
    const int e = idx >> 2;
    const int h = idx & 3;
    const int d = (e < EE) ? dst[e] : (e - EE);
    const float m  = deord(maxb[(size_t)d * NHEADS + h]);
    const float ex = expf(ebuf[idx] - m);
    ebuf[idx] = ex;
    atomicAdd(&den[(size_t)d * NHEADS + h], ex);
}

// ---------------- weighted scatter: aggr[dst] += alpha * xr[src] (wave per edge) --------
__global__ void edge_scatter_kernel(const int* __restrict__ src,
                                    const int* __restrict__ dst,
                                    const float* __restrict__ xr,
                                    const float* __restrict__ ebuf,
                                    const float* __restrict__ den,
                                    float* __restrict__ aggr) {
    const int e = blockIdx.x * (blockDim.x >> 5) + (threadIdx.x >> 5);
    if (e >= ETOT) return;
    const int lane = threadIdx.x & 31;
    int s, d;
    if (e < EE) { s = src[e]; d = dst[e]; } else { s = d = e - EE; }

    const float* pr = xr + (size_t)s * HC;
    float* pa = aggr + (size_t)d * HC;
#pragma unroll
    for (int j = 0; j < NHEADS; ++j) {
        const float alpha = ebuf[(size_t)e * NHEADS + j] /
                            (den[(size_t)d * NHEADS + j] + 1e-16f);
        const int c = lane + 32 * j;                      // head j, channel lane
        atomicAdd(&pa[c], alpha * pr[c]);
    }
}

// ---------------- bias + LayerNorm + GELU + residual (wave per node) ----------------
__global__ void post_ln_kernel(const float* __restrict__ aggr,
                               const float* __restrict__ bo,
                               const float* __restrict__ g,
                               const float* __restrict__ be,
                               float* __restrict__ h,    // read (residual) + write
                               int residual) {
    const int node = blockIdx.x * (blockDim.x >> 5) + (threadIdx.x >> 5);
    if (node >= NN) return;
    const int lane = threadIdx.x & 31;
    const float* pa = aggr + (size_t)node * HC;
    float* ph = h + (size_t)node * HC;

    float v[4];
    float s = 0.0f;
#pragma unroll
    for (int j = 0; j < 4; ++j) {
        const int c = lane + 32 * j;
        v[j] = pa[c] + bo[c];
        s += v[j];
    }
#pragma unroll
    for (int off = 16; off > 0; off >>= 1) s += __shfl_xor(s, off, 32);
    const float mu = s * (1.0f / HC);

    float q = 0.0f;
#pragma unroll
    for (int j = 0; j < 4; ++j) { const float dv = v[j] - mu; q += dv * dv; }
#pragma unroll
    for (int off = 16; off > 0; off >>= 1) q += __shfl_xor(q, off, 32);
    const float inv = rsqrtf(q * (1.0f / HC) + 1e-5f);

#pragma unroll
    for (int j = 0; j < 4; ++j) {
        const int c = lane + 32 * j;
        float t = g[c] * (v[j] - mu) * inv + be[c];
        t = gelu_exact(t);
        ph[c] = residual ? (ph[c] + t) : t;
    }
}

// ---------------- row L2 normalize (wave per node) ----------------
__global__ void l2norm_kernel(float* __restrict__ out) {
    const int node = blockIdx.x * (blockDim.x >> 5) + (threadIdx.x >> 5);
    if (node >= NN) return;
    const int lane = threadIdx.x & 31;
    float* po = out + (size_t)node * EMB;
    float v[4];
    float s = 0.0f;
#pragma unroll
    for (int j = 0; j < 4; ++j) { v[j] = po[lane + 32 * j]; s += v[j] * v[j]; }
#pragma unroll
    for (int off = 16; off > 0; off >>= 1) s += __shfl_xor(s, off, 32);
    const float inv = 1.0f / fmaxf(sqrtf(s), 1e-12f);
#pragma unroll
    for (int j = 0; j < 4; ++j) po[lane + 32 * j] = v[j] * inv;
}

// =====================================================================
extern "C" void kernel_launch(void* const* d_in, const int* in_sizes, int n_in,
                              void* d_out, int out_size, void* d_ws, size_t ws_size,
                              hipStream_t stream) {
    const float* x     = (const float*)d_in[0];
    const int*   ei    = (const int*)d_in[1];
    const int*   src   = ei;
    const int*   dst   = ei + EE;
    const float* W_in  = (const float*)d_in[2];
    const float* b_in  = (const float*)d_in[3];
    const float* W_out = (const float*)d_in[28];
    const float* b_out = (const float*)d_in[29];

    // ---- workspace layout (floats) ----
    float* ws   = (float*)d_ws;
    float* h    = ws;                                  // [NN, HC]
    float* aggr = h    + (size_t)NN * HC;              // [NN, HC]
    float* xl   = aggr + (size_t)NN * HC;              // [NN, HC]
    float* xr   = xl   + (size_t)NN * HC;              // [NN, HC]
    float* ebuf = xr   + (size_t)NN * HC;              // [ETOT, NHEADS]
    unsigned* maxb = (unsigned*)(ebuf + (size_t)ETOT * NHEADS);  // [NN, NHEADS]
    float* den  = (float*)maxb + (size_t)NN * NHEADS;  // [NN, NHEADS]

    const int MTILES = NN / 16;                        // 3125 (exact)

    // 1) h = gelu(x @ W_in + b_in)     [NN,128]@[128,32]
    gemm_wmma_kernel<<<MTILES, 32 * (HIDC / 16), 0, stream>>>(
        x, W_in, b_in, h, NN, INDIM, HIDC, /*act=*/1);

    const int EW_GRID  = (ETOT + 7) / 8;               // wave-per-edge, 8 waves/block
    const int EH_GRID  = (ETOT * NHEADS + 255) / 256;
    const int NW_GRID  = (NN + 7) / 8;

    for (int layer = 0; layer < 3; ++layer) {
        const int base = 4 + 8 * layer;
        const float* Wl  = (const float*)d_in[base + 0];
        const float* bl  = (const float*)d_in[base + 1];
        const float* Wr  = (const float*)d_in[base + 2];
        const float* br  = (const float*)d_in[base + 3];
        const float* att = (const float*)d_in[base + 4];
        const float* bo  = (const float*)d_in[base + 5];
        const float* gg  = (const float*)d_in[base + 6];
        const float* be  = (const float*)d_in[base + 7];
        const int K = (layer == 0) ? HIDC : HC;

        // a) per-node transforms
        gemm_wmma_kernel<<<MTILES, 32 * (HC / 16), 0, stream>>>(
            h, Wl, bl, xl, NN, K, HC, /*act=*/0);
        gemm_wmma_kernel<<<MTILES, 32 * (HC / 16), 0, stream>>>(
            h, Wr, br, xr, NN, K, HC, /*act=*/0);

        // b) zero aggr, maxb (+den contiguous)
        {
            const int n1 = NN * HC;
            fillz_kernel<<<(n1 + 255) / 256, 256, 0, stream>>>(aggr, n1);
            const int n2 = NN * NHEADS * 2;            // maxb + den, adjacent
            fillz_kernel<<<(n2 + 255) / 256, 256, 0, stream>>>((float*)maxb, n2);
        }

        // c) logits + segment max
        edge_logit_kernel<<<EW_GRID, 256, 0, stream>>>(
            src, dst, xl, xr, att, ebuf, maxb);

        // d) exp + segment sum
        edge_exp_kernel<<<EH_GRID, 256, 0, stream>>>(dst, maxb, ebuf, den);

        // e) weighted scatter
        edge_scatter_kernel<<<EW_GRID, 256, 0, stream>>>(
            src, dst, xr, ebuf, den, aggr);

        // f) bias + LN + GELU + residual -> h
        post_ln_kernel<<<NW_GRID, 256, 0, stream>>>(
            aggr, bo, gg, be, h, /*residual=*/(layer > 0) ? 1 : 0);
    }

    // 3) out = h @ W_out + b_out, then row-wise L2 normalize
    gemm_wmma_kernel<<<MTILES, 32 * (EMB / 16), 0, stream>>>(
        h, W_out, b_out, (float*)d_out, NN, HC, EMB, /*act=*/0);
    l2norm_kernel<<<NW_GRID, 256, 0, stream>>>((float*)d_out);
}